// HistogramLayer_25563645346324
// MI455X (gfx1250) — compile-verified
//
#include <hip/hip_runtime.h>

// Histogram params from the reference
#define XMIN   (-3.0f)
#define NCH    64
#define NBINS  256
#define HIST_N (NCH * NBINS)   // 16384 bins -> 64KB of u32 in LDS

typedef float v4f __attribute__((ext_vector_type(4)));
typedef int   v4i __attribute__((ext_vector_type(4)));

#if __has_builtin(__builtin_amdgcn_global_load_async_to_lds_b128) && \
    __has_builtin(__builtin_amdgcn_global_store_async_from_lds_b128) && \
    __has_builtin(__builtin_amdgcn_s_wait_asynccnt)
#define USE_ASYNC_LDS 1
#else
#define USE_ASYNC_LDS 0
#endif

#if USE_ASYNC_LDS
// Per hipcc diagnostics, the builtin signature is:
//   (int4 __device__*, int4 __shared__*, imm int offset, imm int cpol)
typedef __attribute__((address_space(1))) v4i* gv4ip;  // global side
typedef __attribute__((address_space(3))) v4i* lv4ip;  // LDS side
#endif

__global__ void zero_hist_kernel(float* __restrict__ hist) {
    int i = blockIdx.x * blockDim.x + threadIdx.x;
    if (i < HIST_N) hist[i] = 0.0f;
}

__device__ __forceinline__ void bin4(const v4f& v, unsigned c0, unsigned* lhist,
                                     float scale) {
    #pragma unroll
    for (int k = 0; k < 4; ++k) {
        // TF histogram_fixed_width semantics: clip-low, truncate, clamp-high
        float t = (v[k] - XMIN) * scale;
        t = fmaxf(t, 0.0f);
        int b = (int)t;
        b = b < (NBINS - 1) ? b : (NBINS - 1);
        unsigned c = c0 + (unsigned)k;
        // XOR bank swizzle: lanes sharing a bin differ in channel -> distinct banks
        unsigned idx = (c << 8) | ((unsigned)b ^ c);
        atomicAdd(&lhist[idx], 1u);     // ds_add_u32
    }
}

__global__ __launch_bounds__(256, 2)
void hist_copy_kernel(const float* __restrict__ x,
                      float* __restrict__ out,
                      float* __restrict__ hist,   // [64*256] f32, pre-zeroed
                      int nvec,                   // number of float4 chunks
                      int tail)                   // leftover scalars (0..3)
{
    // 64KB privatized histogram per workgroup.
    __shared__ unsigned int lhist[HIST_N];
#if USE_ASYNC_LDS
    // Double-buffered DMA staging: 2 x 256 lanes x 16B = 8KB.
    __shared__ v4f stage[2][256];
#endif
    #pragma unroll 8
    for (int i = threadIdx.x; i < HIST_N; i += 256) lhist[i] = 0u;
    __syncthreads();

    const float scale = (float)NBINS / 6.0f;  // nbins / (xmax - xmin)
    const int tid    = threadIdx.x;
    const int gid    = blockIdx.x * 256 + tid;
    const int stride = gridDim.x * 256;
    const v4f* __restrict__ xv = (const v4f*)x;
    v4f* __restrict__ ov       = (v4f*)out;

#if USE_ASYNC_LDS
    // Async DMA pipeline: global -> LDS (load), LDS -> VGPR (binning),
    // LDS -> global (store) without the copy data ever living in VGPRs
    // for the store side. One s_wait_asynccnt 0 per iteration guarantees:
    //   (a) this tile's load has landed in stage[par]
    //   (b) the store issued from stage[par^1] last iteration completed,
    //       so it is safe to overwrite with the next prefetch load.
    int par = 0;
    if (gid < nvec) {
        __builtin_amdgcn_global_load_async_to_lds_b128(
            (gv4ip)&xv[gid], (lv4ip)&stage[0][tid], 0, 0);
    }
    for (int i = gid; i < nvec; i += stride, par ^= 1) {
        __builtin_amdgcn_s_wait_asynccnt(0);
        int inext = i + stride;
        if (inext < nvec) {
            __builtin_amdgcn_global_load_async_to_lds_b128(
                (gv4ip)&xv[inext], (lv4ip)&stage[par ^ 1][tid], 0, 0);
        }
        v4f v = stage[par][tid];                       // ds_load_b128
        __builtin_amdgcn_global_store_async_from_lds_b128(
            (gv4ip)&ov[i], (lv4ip)&stage[par][tid], 0, 0);
        bin4(v, (unsigned)((i << 2) & 63), lhist, scale);  // overlaps DMA
    }
    __builtin_amdgcn_s_wait_asynccnt(0);  // drain before LDS flush reuse
#else
    for (int i = gid; i < nvec; i += stride) {
        // Streamed once each way; NT hints keep the 192MB L2 unpolluted.
        v4f v = __builtin_nontemporal_load(&xv[i]);
        __builtin_nontemporal_store(v, &ov[i]);
        bin4(v, (unsigned)((i << 2) & 63), lhist, scale);
    }
#endif

    // Scalar tail (N % 4), handled by block 0 before the flush barrier.
    if (tail && blockIdx.x == 0 && (int)threadIdx.x < tail) {
        int e = nvec * 4 + threadIdx.x;
        float v = x[e];
        out[e] = v;
        float t = (v - XMIN) * scale;
        t = fmaxf(t, 0.0f);
        int b = (int)t;
        b = b < (NBINS - 1) ? b : (NBINS - 1);
        unsigned c = (unsigned)(e & 63);
        atomicAdd(&lhist[(c << 8) | ((unsigned)b ^ c)], 1u);
    }
    __syncthreads();

    // Flush: un-swizzle and accumulate with hardware f32 global atomics.
    // Counts are integers < 2^24, so f32 accumulation is exact.
    for (int p = threadIdx.x; p < HIST_N; p += 256) {
        unsigned cnt = lhist[p];
        if (cnt) {
            unsigned c = (unsigned)p >> 8;
            unsigned b = ((unsigned)p & 255u) ^ c;
            unsafeAtomicAdd(&hist[(c << 8) | b], (float)cnt); // global_atomic_add_f32
        }
    }
}

extern "C" void kernel_launch(void* const* d_in, const int* in_sizes, int n_in,
                              void* d_out, int out_size, void* d_ws, size_t ws_size,
                              hipStream_t stream) {
    (void)n_in; (void)d_ws; (void)ws_size; (void)out_size;

    const float* x = (const float*)d_in[0];
    int N = in_sizes[0];                          // 64*112*112*64 = 51,380,224
    float* out  = (float*)d_out;                  // [N] pass-through
    float* hist = out + N;                        // [64*256] histogram (f32)

    // 1) Zero the histogram region (harness poisons d_out).
    zero_hist_kernel<<<(HIST_N + 255) / 256, 256, 0, stream>>>(hist);

    // 2) Fused streaming copy + privatized histogram via async LDS DMA.
    int nvec = N >> 2;
    int tail = N & 3;
    int want = (nvec + 255) / 256;
    int blocks = want < 512 ? (want > 0 ? want : 1) : 512;
    hist_copy_kernel<<<blocks, 256, 0, stream>>>(x, out, hist, nvec, tail);
}